// GATWithSkips_29283087024638
// MI455X (gfx1250) — compile-verified
//
#include <hip/hip_runtime.h>
#include <hip/hip_bf16.h>
#include <math.h>

// GATWithSkips for MI455X (gfx1250).
// GEMMs use V_WMMA_F32_16X16X4_F32 (full fp32 precision; 82 GFLOP total is
// memory-roofline-bound at 23.3 TB/s, so fp32 WMMA is the right precision).
// Weights are pre-packed row-pair-interleaved so each B fragment is a single
// coalesced b64 load; each wave computes a 16x(16*NT) C strip so one A
// fragment feeds NT WMMAs. Segment softmax + scatter-add use agent-scope
// relaxed HW atomics.

typedef float v2f __attribute__((ext_vector_type(2)));
typedef float v8f __attribute__((ext_vector_type(8)));

__device__ __forceinline__ float lrelu(float v) { return v > 0.f ? v : 0.2f * v; }

__device__ __forceinline__ void atomicMaxF32(float* addr, float v) {
  // int-trick float max: works with -inf init (0xFF800000).
  if (v >= 0.f) {
    (void)__hip_atomic_fetch_max((int*)addr, __float_as_int(v),
                                 __ATOMIC_RELAXED, __HIP_MEMORY_SCOPE_AGENT);
  } else {
    (void)__hip_atomic_fetch_min((unsigned int*)addr, __float_as_uint(v),
                                 __ATOMIC_RELAXED, __HIP_MEMORY_SCOPE_AGENT);
  }
}

__device__ __forceinline__ void atomicAddF32(float* addr, float v) {
  (void)__hip_atomic_fetch_add(addr, v, __ATOMIC_RELAXED, __HIP_MEMORY_SCOPE_AGENT);
}

// Pack B[K,Nc] (row-major) into Bp[r][n][2] = {B[2r][n], B[2r+1][n]}.
// A WMMA B-fragment always consumes the even/odd row pair (2r, 2r+1), so the
// packed layout turns the per-lane fragment into one aligned b64 load.
__global__ void gat_pack_b(const float* __restrict__ B, float* __restrict__ Bp,
                           int K, int Nc)
{
  unsigned i = blockIdx.x * blockDim.x + threadIdx.x;
  unsigned total = (unsigned)(K / 2) * (unsigned)Nc;
  if (i >= total) return;
  unsigned r = i / Nc, n = i - r * Nc;
  Bp[2 * (size_t)i]     = B[(size_t)(2 * r)     * Nc + n];
  Bp[2 * (size_t)i + 1] = B[(size_t)(2 * r + 1) * Nc + n];
}

// ---------------------------------------------------------------------------
// C[M,Nc] = A[M,K] @ B[K,Nc] (+bias | +=), B given packed (see gat_pack_b).
// One wave per 16x(16*NT) C strip.
// A-frag  (16x4 f32):  lane l: m=l&15, holds A[m][koff], A[m][koff+1], koff=(l>>4)*2
// B-frag  (4x16 f32):  lane l: n=l&15, holds B[koff][n], B[koff+1][n]
// C-frag  (16x16 f32): VGPR v: row v + (l>>4)*8, col l&15
// Requires M%16==0, Nc%(16*NT)==0, K%4==0 (true for all calls here).
// ---------------------------------------------------------------------------
template <int NT>
__global__ __launch_bounds__(256) void gat_gemm_wmma_f32(
    const float* __restrict__ A, const float* __restrict__ Bp,
    const float* __restrict__ bias, float* __restrict__ C,
    int M, int K, int Nc, int accumulate)
{
  const int lane = threadIdx.x & 31;
  const int wave = threadIdx.x >> 5;
  const int stripsN = (Nc >> 4) / NT;
  const int totalStrips = (M >> 4) * stripsN;
  const int strip = blockIdx.x * (blockDim.x >> 5) + wave;
  if (strip >= totalStrips) return;        // wave-uniform: EXEC stays all-ones
  const int tm = strip / stripsN;
  const int tn0 = (strip - tm * stripsN) * NT;
  const int mi = lane & 15;
  const int koff = (lane >> 4) << 1;       // 0 or 2
  const int col0 = tn0 * 16 + mi;

  const float* __restrict__ ap = A + (size_t)(tm * 16 + mi) * K + koff;
  // fragment row-pair index r = k/2 + koff/2; packed row stride = 2*Nc floats
  const float* __restrict__ bp = Bp + ((size_t)(koff >> 1) * Nc + col0) * 2;
  const size_t bstep = (size_t)4 * Nc;     // 2 row-pairs per k-step of 4

  v8f acc[NT];
#pragma unroll
  for (int t = 0; t < NT; ++t) acc[t] = (v8f){0.f, 0.f, 0.f, 0.f, 0.f, 0.f, 0.f, 0.f};

#pragma unroll 2
  for (int k = 0; k < K; k += 4) {
    v2f a = *(const v2f*)ap;
    v2f b[NT];
#pragma unroll
    for (int t = 0; t < NT; ++t) b[t] = *(const v2f*)(bp + 32 * t);
#pragma unroll
    for (int t = 0; t < NT; ++t)
      acc[t] = __builtin_amdgcn_wmma_f32_16x16x4_f32(false, a, false, b[t],
                                                     (short)0, acc[t], false, false);
    ap += 4;
    bp += bstep;
  }

  const int rbase = tm * 16 + ((lane >> 4) << 3);
#pragma unroll
  for (int t = 0; t < NT; ++t) {
    const int col = col0 + 16 * t;
    const float bv = bias ? bias[col] : 0.f;
#pragma unroll
    for (int v = 0; v < 8; ++v) {
      size_t o = (size_t)(rbase + v) * Nc + col;
      float val = acc[t][v];
      C[o] = accumulate ? (C[o] + val) : (val + bv);
    }
  }
}

// a_src[i] = h[i,:] . att_src[h], a_dst[i] = h[i,:] . att_dst[h]  (i = n*H+h)
__global__ void gat_att_scores(const float* __restrict__ h,
                               const float* __restrict__ att_s,
                               const float* __restrict__ att_d,
                               float* __restrict__ osrc, float* __restrict__ odst,
                               int NH, int H, int Cc)
{
  int idx = blockIdx.x * blockDim.x + threadIdx.x;
  if (idx >= NH) return;
  int hh = idx % H;
  const float* __restrict__ hp = h + (size_t)idx * Cc;
  const float* __restrict__ as = att_s + hh * Cc;
  const float* __restrict__ ad = att_d + hh * Cc;
  float s = 0.f, d = 0.f;
  for (int c = 0; c < Cc; ++c) { float v = hp[c]; s += v * as[c]; d += v * ad[c]; }
  osrc[idx] = s;
  odst[idx] = d;
}

// Pass 1: per-destination segment max of leaky_relu(a_src[s]+a_dst[d])
__global__ void gat_edge_max(const int* __restrict__ srcE, const int* __restrict__ dstE,
                             const float* __restrict__ asrc, const float* __restrict__ adst,
                             float* __restrict__ emax, int E, int Nn, int H)
{
  unsigned idx = blockIdx.x * blockDim.x + threadIdx.x;
  unsigned total = (unsigned)(E + Nn) * (unsigned)H;
  if (idx >= total) return;
  unsigned e = idx / H, hh = idx - e * H;   // edges [0,E) real, [E,E+N) self loops
  int s = (e < (unsigned)E) ? srcE[e] : (int)(e - E);
  int d = (e < (unsigned)E) ? dstE[e] : (int)(e - E);
  float v = lrelu(asrc[(size_t)s * H + hh] + adst[(size_t)d * H + hh]);
  atomicMaxF32(&emax[(size_t)d * H + hh], v);
}

// Pass 2: w = exp(logit - emax[d]); store w; denom[d] += w
__global__ void gat_edge_expsum(const int* __restrict__ srcE, const int* __restrict__ dstE,
                                const float* __restrict__ asrc, const float* __restrict__ adst,
                                const float* __restrict__ emax,
                                float* __restrict__ wbuf, float* __restrict__ denom,
                                int E, int Nn, int H)
{
  unsigned idx = blockIdx.x * blockDim.x + threadIdx.x;
  unsigned total = (unsigned)(E + Nn) * (unsigned)H;
  if (idx >= total) return;
  unsigned e = idx / H, hh = idx - e * H;
  int s = (e < (unsigned)E) ? srcE[e] : (int)(e - E);
  int d = (e < (unsigned)E) ? dstE[e] : (int)(e - E);
  float v = lrelu(asrc[(size_t)s * H + hh] + adst[(size_t)d * H + hh]);
  float w = expf(v - emax[(size_t)d * H + hh]);
  wbuf[idx] = w;
  atomicAddF32(&denom[(size_t)d * H + hh], w);
}

// Pass 3: out[d, :] += h[s, :] * (w / (denom[d]+1e-16)); one thread per (edge, channel)
__global__ void gat_edge_aggregate(const int* __restrict__ srcE, const int* __restrict__ dstE,
                                   const float* __restrict__ h,
                                   const float* __restrict__ wbuf,
                                   const float* __restrict__ denom,
                                   float* __restrict__ out,
                                   int E, int Nn, int H, int Cc)
{
  unsigned idx = blockIdx.x * blockDim.x + threadIdx.x;
  unsigned HC = (unsigned)H * (unsigned)Cc;
  unsigned total = (unsigned)(E + Nn) * HC;   // <= 460.8M, fits unsigned
  if (idx >= total) return;
  unsigned e = idx / HC, k = idx - e * HC;
  int s = (e < (unsigned)E) ? srcE[e] : (int)(e - E);
  int d = (e < (unsigned)E) ? dstE[e] : (int)(e - E);
  unsigned hh = k / Cc;
  float alpha = wbuf[(size_t)e * H + hh] / (denom[(size_t)d * H + hh] + 1e-16f);
  atomicAddF32(&out[(size_t)d * HC + k], h[(size_t)s * HC + k] * alpha);
}

__global__ void gat_fill(float* __restrict__ p, float v, unsigned n)
{
  unsigned i = blockIdx.x * blockDim.x + threadIdx.x;
  if (i < n) p[i] = v;
}

__global__ void gat_bias_add(float* __restrict__ p, const float* __restrict__ b,
                             unsigned total, int ncols)
{
  unsigned i = blockIdx.x * blockDim.x + threadIdx.x;
  if (i < total) p[i] += b[i % ncols];
}

static inline unsigned cdivu(size_t a, size_t b) { return (unsigned)((a + b - 1) / b); }

extern "C" void kernel_launch(void* const* d_in, const int* in_sizes, int n_in,
                              void* d_out, int out_size, void* d_ws, size_t ws_size,
                              hipStream_t stream)
{
  (void)n_in; (void)out_size; (void)ws_size;
  const float* x     = (const float*)d_in[0];
  const int*   ei    = (const int*)d_in[1];
  const float* g1_W  = (const float*)d_in[2];
  const float* g1_as = (const float*)d_in[3];
  const float* g1_ad = (const float*)d_in[4];
  const float* g1_b  = (const float*)d_in[5];
  const float* sk1_W = (const float*)d_in[6];
  const float* sk1_b = (const float*)d_in[7];
  const float* sk2_W = (const float*)d_in[8];
  const float* sk2_b = (const float*)d_in[9];
  const float* g3_W  = (const float*)d_in[10];
  const float* g3_as = (const float*)d_in[11];
  const float* g3_ad = (const float*)d_in[12];
  const float* g3_b  = (const float*)d_in[13];
  float* out = (float*)d_out;

  const int IN = 128, H1 = 8, C1 = 64, HC1 = 512, NC = 16;
  const int Nn = in_sizes[0] / IN;
  const int E  = in_sizes[1] / 2;
  const int* srcE = ei;       // edge_index[0]
  const int* dstE = ei + E;   // edge_index[1]
  const int ET = E + Nn;      // with self loops

  // -------- workspace layout (256B aligned slabs) --------
  char* ws = (char*)d_ws;
  size_t ofs = 0;
  auto alloc = [&](size_t nfloats) -> float* {
    float* p = (float*)(ws + ofs);
    ofs += ((nfloats * sizeof(float) + 255) / 256) * 256;
    return p;
  };
  float* h1     = alloc((size_t)Nn * HC1);   // h = x@g1_W ; reused later as xs2
  float* x1     = alloc((size_t)Nn * HC1);   // gat1 output
  float* xs1    = alloc((size_t)Nn * HC1);   // skip1
  float* asrc1  = alloc((size_t)Nn * H1);
  float* adst1  = alloc((size_t)Nn * H1);
  float* emax1  = alloc((size_t)Nn * H1);
  float* denom1 = alloc((size_t)Nn * H1);
  float* wbuf1  = alloc((size_t)ET * H1);
  float* h3     = alloc((size_t)Nn * NC);
  float* as3    = alloc((size_t)Nn);
  float* ad3    = alloc((size_t)Nn);
  float* emax3  = alloc((size_t)Nn);
  float* denom3 = alloc((size_t)Nn);
  float* wbuf3  = alloc((size_t)ET);
  float* g1_Wp  = alloc((size_t)IN  * HC1);  // packed weights
  float* sk1_Wp = alloc((size_t)IN  * HC1);
  float* sk2_Wp = alloc((size_t)HC1 * HC1);
  float* g3_Wp  = alloc((size_t)2 * HC1 * NC);
  float* xs2    = h1;   // h1 is dead after layer-1 aggregation

  const float NEG_INF = -__builtin_huge_valf();
  const int BT = 256;

  // -------- pack weights (row-pair interleave) --------
  gat_pack_b<<<cdivu((size_t)(IN / 2) * HC1, BT), BT, 0, stream>>>(g1_W, g1_Wp, IN, HC1);
  gat_pack_b<<<cdivu((size_t)(IN / 2) * HC1, BT), BT, 0, stream>>>(sk1_W, sk1_Wp, IN, HC1);
  gat_pack_b<<<cdivu((size_t)(HC1 / 2) * HC1, BT), BT, 0, stream>>>(sk2_W, sk2_Wp, HC1, HC1);
  gat_pack_b<<<cdivu((size_t)HC1 * NC, BT), BT, 0, stream>>>(g3_W, g3_Wp, 2 * HC1, NC);

  // -------- init accumulators --------
  gat_fill<<<cdivu((size_t)Nn * H1, BT), BT, 0, stream>>>(emax1,  NEG_INF, (unsigned)(Nn * H1));
  gat_fill<<<cdivu((size_t)Nn * H1, BT), BT, 0, stream>>>(denom1, 0.f,     (unsigned)(Nn * H1));
  gat_fill<<<cdivu((size_t)Nn * HC1, BT), BT, 0, stream>>>(x1,    0.f,     (unsigned)((size_t)Nn * HC1));
  gat_fill<<<cdivu((size_t)Nn, BT), BT, 0, stream>>>(emax3,  NEG_INF, (unsigned)Nn);
  gat_fill<<<cdivu((size_t)Nn, BT), BT, 0, stream>>>(denom3, 0.f,     (unsigned)Nn);
  gat_fill<<<cdivu((size_t)Nn * NC, BT), BT, 0, stream>>>(out, 0.f, (unsigned)(Nn * NC));

  // -------- layer 1 GEMMs (WMMA fp32): h1 = x@g1_W ; xs1 = x@sk1_W + b --------
  const unsigned strips512 = cdivu((size_t)(Nn / 16) * (HC1 / 64), 8);
  gat_gemm_wmma_f32<4><<<strips512, BT, 0, stream>>>(x, g1_Wp, nullptr, h1, Nn, IN, HC1, 0);
  gat_gemm_wmma_f32<4><<<strips512, BT, 0, stream>>>(x, sk1_Wp, sk1_b, xs1, Nn, IN, HC1, 0);

  // -------- layer 1 attention + segment softmax + aggregate --------
  gat_att_scores<<<cdivu((size_t)Nn * H1, BT), BT, 0, stream>>>(h1, g1_as, g1_ad,
                                                                asrc1, adst1, Nn * H1, H1, C1);
  const unsigned eh1 = cdivu((size_t)ET * H1, BT);
  gat_edge_max<<<eh1, BT, 0, stream>>>(srcE, dstE, asrc1, adst1, emax1, E, Nn, H1);
  gat_edge_expsum<<<eh1, BT, 0, stream>>>(srcE, dstE, asrc1, adst1, emax1, wbuf1, denom1, E, Nn, H1);
  gat_edge_aggregate<<<cdivu((size_t)ET * HC1, BT), BT, 0, stream>>>(srcE, dstE, h1, wbuf1,
                                                                     denom1, x1, E, Nn, H1, C1);
  gat_bias_add<<<cdivu((size_t)Nn * HC1, BT), BT, 0, stream>>>(x1, g1_b,
                                                               (unsigned)((size_t)Nn * HC1), HC1);

  // -------- skip2 + layer 2 input GEMMs --------
  // xs2 = x1 @ sk2_W + b      (reuses h1 buffer)
  gat_gemm_wmma_f32<4><<<strips512, BT, 0, stream>>>(x1, sk2_Wp, sk2_b, xs2, Nn, HC1, HC1, 0);
  // h3 = [xs1 | xs2] @ g3_W  == xs1@g3_W[:512] + xs2@g3_W[512:]
  const unsigned strips16 = cdivu((size_t)(Nn / 16) * (NC / 16), 8);
  gat_gemm_wmma_f32<1><<<strips16, BT, 0, stream>>>(xs1, g3_Wp, nullptr, h3, Nn, HC1, NC, 0);
  gat_gemm_wmma_f32<1><<<strips16, BT, 0, stream>>>(xs2, g3_Wp + (size_t)HC1 * NC, nullptr, h3,
                                                    Nn, HC1, NC, 1);

  // -------- layer 2 (1 head, 16 ch) --------
  gat_att_scores<<<cdivu((size_t)Nn, BT), BT, 0, stream>>>(h3, g3_as, g3_ad, as3, ad3, Nn, 1, NC);
  const unsigned eh3 = cdivu((size_t)ET, BT);
  gat_edge_max<<<eh3, BT, 0, stream>>>(srcE, dstE, as3, ad3, emax3, E, Nn, 1);
  gat_edge_expsum<<<eh3, BT, 0, stream>>>(srcE, dstE, as3, ad3, emax3, wbuf3, denom3, E, Nn, 1);
  gat_edge_aggregate<<<cdivu((size_t)ET * NC, BT), BT, 0, stream>>>(srcE, dstE, h3, wbuf3,
                                                                    denom3, out, E, Nn, 1, NC);
  gat_bias_add<<<cdivu((size_t)Nn * NC, BT), BT, 0, stream>>>(out, g3_b, (unsigned)(Nn * NC), NC);
}